// ValueModel_10823317586373
// MI455X (gfx1250) — compile-verified
//
#include <hip/hip_runtime.h>

typedef __attribute__((ext_vector_type(2))) float v2f;
typedef __attribute__((ext_vector_type(8))) float v8f;

#define INPUT_SIZE 16
#define KN         50      // neurons added per layer
#define HC         25      // hidden layers
#define FW         1266    // final width = 16 + 25*50
#define ROWS       64      // batch rows per workgroup
#define XSTRIDE    1268    // even (b64 align); 1268%64=52 -> conflict-free A loads
#define NTHREADS   512     // 16 waves = 4 N-tiles x 2 M-pairs x 2 K-halves

__global__ __launch_bounds__(NTHREADS) void densenet_wmma_kernel(
    const float* __restrict__ state,   // (B, 16)
    const float* __restrict__ Ws,      // (25, 50, 1266)
    const float* __restrict__ bs,      // (25, 50)
    const float* __restrict__ Wout,    // (1266, 1)
    const float* __restrict__ bout,    // (1,)
    float* __restrict__ out)           // (B, 1)
{
    extern __shared__ float xbuf[];    // ROWS x XSTRIDE fp32 activation slab

    const int tid  = threadIdx.x;
    const int wave = tid >> 5;
    const int lane = tid & 31;
    const int l16  = lane & 15;
    const int hi   = lane >> 4;        // 0: K sub 0,1   1: K sub 2,3
    const int ksub = hi * 2;

    const int nt = wave & 3;           // N tile (neurons 16*nt..16*nt+15)
    const int mp = (wave >> 2) & 1;    // M pair: rows mp*32 .. mp*32+31
    const int kh = wave >> 3;          // K half: 0 = low (carries bias), 1 = high(+tail)

    // wave-uniform flags as scalars
    const int kh_u   = __builtin_amdgcn_readfirstlane(kh);
    const int rowbase = blockIdx.x * ROWS;

    // ---- stage initial 64x16 state tile into LDS ----
    for (int e = tid; e < ROWS * INPUT_SIZE; e += NTHREADS) {
        int r = e >> 4, c = e & 15;
        xbuf[r * XSTRIDE + c] = state[(size_t)(rowbase + r) * INPUT_SIZE + c];
    }
    __syncthreads();

    const int  n      = nt * 16 + l16;          // neuron this lane covers
    const bool nvalid = (n < KN);
    const int  ncl    = nvalid ? n : (KN - 1);  // clamped: garbage lanes compute
                                                // discarded columns safely
    const float* xrow0 = xbuf + (mp * 32 + l16) * XSTRIDE;   // A-frag, M tile 0
    const float* xrow1 = xrow0 + 16 * XSTRIDE;               // A-frag, M tile 1

    int width = INPUT_SIZE;
    for (int layer = 0; layer < HC; ++layer) {
        const float* wrow = Ws + (size_t)(layer * KN + ncl) * FW;

        const float bias = (kh_u == 0) ? bs[layer * KN + ncl] : 0.0f;
        v8f acc0, acc1;
        #pragma unroll
        for (int j = 0; j < 8; ++j) { acc0[j] = bias; acc1[j] = bias; }

        const int wmain  = width & ~3;
        const int ksplit = (wmain >> 1) & ~3;
        // wave-uniform K range -> force to SGPRs so the loop gets scalar
        // control flow (and the unroll/clause structure) instead of
        // exec-masked vector loop control.
        const int kbeg = __builtin_amdgcn_readfirstlane((kh_u == 0) ? 0 : ksplit);
        const int kend = __builtin_amdgcn_readfirstlane((kh_u == 0) ? ksplit : wmain);

        #pragma unroll 4
        for (int k = kbeg; k < kend; k += 4) {
            v2f b  = *(const v2f*)(wrow  + k + ksub);   // 1 global b64 ...
            v2f a0 = *(const v2f*)(xrow0 + k + ksub);
            v2f a1 = *(const v2f*)(xrow1 + k + ksub);
            acc0 = __builtin_amdgcn_wmma_f32_16x16x4_f32(   // ... feeds 2 WMMAs
                       false, a0, false, b, (short)0, acc0, false, false);
            acc1 = __builtin_amdgcn_wmma_f32_16x16x4_f32(
                       false, a1, false, b, (short)0, acc1, false, false);
        }
        if ((kh_u == 1) && (width & 2)) { // K tail of 2: zero upper half (no NaNs)
            v2f a0, a1, b;
            if (!hi) {
                a0 = *(const v2f*)(xrow0 + wmain);
                a1 = *(const v2f*)(xrow1 + wmain);
                b  = *(const v2f*)(wrow  + wmain);
            } else {
                a0[0] = 0.0f; a0[1] = 0.0f;
                a1[0] = 0.0f; a1[1] = 0.0f;
                b[0]  = 0.0f; b[1]  = 0.0f;
            }
            acc0 = __builtin_amdgcn_wmma_f32_16x16x4_f32(
                       false, a0, false, b, (short)0, acc0, false, false);
            acc1 = __builtin_amdgcn_wmma_f32_16x16x4_f32(
                       false, a1, false, b, (short)0, acc1, false, false);
        }

        // phase 1: high-K waves scatter raw partials into the (dead) dest columns
        if (kh_u == 1 && nvalid) {
            float* dst = xbuf + width + n;
            #pragma unroll
            for (int j = 0; j < 8; ++j) {
                int r0 = mp * 32 + hi * 8 + j;            // D-fragment row map
                dst[r0 * XSTRIDE]        = acc0[j];
                dst[(r0 + 16) * XSTRIDE] = acc1[j];
            }
        }
        __syncthreads();
        // phase 2: low-K waves (bias already in acc) combine + leaky-ReLU + store
        if (kh_u == 0 && nvalid) {
            float* dst = xbuf + width + n;
            #pragma unroll
            for (int j = 0; j < 8; ++j) {
                int r0 = mp * 32 + hi * 8 + j;
                float v0 = acc0[j] + dst[r0 * XSTRIDE];
                float v1 = acc1[j] + dst[(r0 + 16) * XSTRIDE];
                v0 = (v0 > 0.0f) ? v0 : 0.01f * v0;
                v1 = (v1 > 0.0f) ? v1 : 0.01f * v1;
                dst[r0 * XSTRIDE]        = v0;
                dst[(r0 + 16) * XSTRIDE] = v1;
            }
        }
        width += KN;
        __syncthreads();
    }

    // ---- final projection: out = x @ Wout + bout (8 threads per row) ----
    const int row = tid >> 3;          // 0..63
    const int g   = tid & 7;
    const float* xr = xbuf + row * XSTRIDE;
    float facc = 0.0f;
    for (int c = g; c < FW; c += 8)
        facc += xr[c] * Wout[c];
    facc += __shfl_down(facc, 4, 8);
    facc += __shfl_down(facc, 2, 8);
    facc += __shfl_down(facc, 1, 8);
    if (g == 0)
        out[rowbase + row] = facc + bout[0];
}

extern "C" void kernel_launch(void* const* d_in, const int* in_sizes, int n_in,
                              void* d_out, int out_size, void* d_ws, size_t ws_size,
                              hipStream_t stream) {
    (void)n_in; (void)d_ws; (void)ws_size; (void)out_size;
    const float* state = (const float*)d_in[0];
    const float* Ws    = (const float*)d_in[1];
    const float* bs    = (const float*)d_in[2];
    const float* Wout  = (const float*)d_in[3];
    const float* bout  = (const float*)d_in[4];
    float* out = (float*)d_out;

    const int B      = in_sizes[0] / INPUT_SIZE;   // 32768
    const int blocks = B / ROWS;                   // 512
    const size_t lds_bytes = (size_t)ROWS * XSTRIDE * sizeof(float); // 324,608 B

    (void)hipFuncSetAttribute((const void*)densenet_wmma_kernel,
                              hipFuncAttributeMaxDynamicSharedMemorySize,
                              (int)lds_bytes);

    densenet_wmma_kernel<<<blocks, NTHREADS, lds_bytes, stream>>>(
        state, Ws, bs, Wout, bout, out);
}